// kmeans_59631325938143
// MI455X (gfx1250) — compile-verified
//
#include <hip/hip_runtime.h>
#include <hip/hip_bf16.h>

// proj = F^T h  (K=64 x D=1024, reduced over N=8192), out = ||h||^2 - ||proj||^2
// fp32 WMMA path: v_wmma_f32_16x16x4_f32, wave32, grid = 64 d-tiles x 4 k-tiles.

#define N_DIM 8192
#define D_DIM 1024
#define K_DIM 64
#define WAVES 8          // waves per block (256 threads)
#define ITERS (N_DIM / (4 * WAVES))   // n-steps of 4 rows per wave = 256

typedef __attribute__((ext_vector_type(2))) float v2f;
typedef __attribute__((ext_vector_type(8))) float v8f;

__global__ __launch_bounds__(256) void ortho_loss_kernel(
    const float* __restrict__ h, const float* __restrict__ F,
    float* __restrict__ out) {
  const int lane = threadIdx.x & 31;
  const int wave = threadIdx.x >> 5;        // 0..7
  const int col  = lane & 15;               // m (A) / d-col (B/C)
  const int half = lane >> 4;               // selects K_in pair {0,1} vs {2,3}

  const int d_tile = blockIdx.x & 63;       // 64 tiles of 16 columns of D
  const int k_tile = blockIdx.x >> 6;       // 4 tiles of 16 rows of K

  // Per-lane base pointers (column of h / column of F this lane reads)
  const float* hp = h + (size_t)(d_tile * 16 + col);   // + n*D_DIM
  const float* Fp = F + (size_t)(k_tile * 16 + col);   // + n*K_DIM

  __shared__ float red[WAVES][256];   // 8 VGPRs x 32 lanes per wave
  __shared__ float s_sum;

  v8f acc0 = {};
  v8f acc1 = {};
  float t1 = 0.f;

  // Each wave covers n = wave*4 + it*32 + {0..3}; all waves run ITERS steps
  // uniformly (EXEC all-1s through every WMMA, per ISA requirement).
  #pragma unroll 2
  for (int it = 0; it < ITERS; it += 2) {
    {
      const int n0 = wave * 4 + it * (4 * WAVES) + 2 * half;
      v2f a, b;
      b.x = hp[(size_t)n0 * D_DIM];
      b.y = hp[(size_t)(n0 + 1) * D_DIM];
      a.x = Fp[(size_t)n0 * K_DIM];
      a.y = Fp[(size_t)(n0 + 1) * K_DIM];
      t1 += b.x * b.x + b.y * b.y;
      acc0 = __builtin_amdgcn_wmma_f32_16x16x4_f32(
          false, a, false, b, (short)0, acc0, false, false);
    }
    {
      const int n0 = wave * 4 + (it + 1) * (4 * WAVES) + 2 * half;
      v2f a, b;
      b.x = hp[(size_t)n0 * D_DIM];
      b.y = hp[(size_t)(n0 + 1) * D_DIM];
      a.x = Fp[(size_t)n0 * K_DIM];
      a.y = Fp[(size_t)(n0 + 1) * K_DIM];
      t1 += b.x * b.x + b.y * b.y;
      acc1 = __builtin_amdgcn_wmma_f32_16x16x4_f32(
          false, a, false, b, (short)0, acc1, false, false);
    }
  }

  // Combine the 8 waves' partial C tiles in LDS before squaring.
  if (threadIdx.x == 0) s_sum = 0.f;
  #pragma unroll
  for (int r = 0; r < 8; ++r)
    red[wave][r * 32 + lane] = acc0[r] + acc1[r];
  __syncthreads();

  float tot = 0.f;
  #pragma unroll
  for (int w = 0; w < WAVES; ++w) tot += red[w][threadIdx.x];

  // h is loaded exactly 4x across the 4 k-splits -> scale t1 by 1/4.
  const float contrib = 0.25f * t1 - tot * tot;
  atomicAdd(&s_sum, contrib);           // ds_add_f32
  __syncthreads();
  if (threadIdx.x == 0) atomicAdd(out, s_sum);   // global_atomic_add_f32
}

extern "C" void kernel_launch(void* const* d_in, const int* in_sizes, int n_in,
                              void* d_out, int out_size, void* d_ws, size_t ws_size,
                              hipStream_t stream) {
  const float* h = (const float*)d_in[0];   // [N, D] fp32
  const float* F = (const float*)d_in[1];   // [N, K] fp32
  float* out = (float*)d_out;               // scalar fp32

  hipMemsetAsync(out, 0, sizeof(float), stream);
  ortho_loss_kernel<<<dim3(256), dim3(256), 0, stream>>>(h, F, out);
}